// MultiScaleHierarchicalFusion_13056700579978
// MI455X (gfx1250) — compile-verified
//
#include <hip/hip_runtime.h>
#include <math.h>

// ---------------- constants from the reference ----------------
#define BATCH   2
#define NIN     1024
#define F       256
#define METAF   256
#define HID     64
#define NT      412      // 205 + 205 + meta + cls
#define KPOOL   205      // ceil(0.2*1024)
#define KPR     42       // ceil(0.1*412)
#define UH      128      // IN_F/2
#define OUTF    128
#define LN_EPS  1e-5f

typedef __attribute__((ext_vector_type(2))) float v2f;
typedef __attribute__((ext_vector_type(8))) float v8f;

__device__ __forceinline__ float act_apply(float v, int act) {
    if (act == 1) return tanhf(v);
    if (act == 2) return fmaxf(v, 0.0f);
    if (act == 3) return v > 0.0f ? v : (expf(v) - 1.0f);      // elu
    if (act == 4) return 1.0f / (1.0f + expf(-v));             // sigmoid
    return v;
}

// ---------------------------------------------------------------------------
// f32 GEMM on the CDNA5 matrix core: C = act(A[M,K] @ B[K,N] + bias)
// 4-wave workgroup computes a 64(M)x16(N) C panel; each wave owns a 16x16
// tile (V_WMMA_F32_16X16X4_F32, K stepped by 4). The 16-wide B column slice
// is staged through LDS in 64-deep K chunks shared by all 4 waves; A streams
// per-lane as b64 loads along K with a one-chunk-ahead global_prefetch_b8.
// Main chunk body is compile-time bounded + unrolled so loads pipeline ahead
// of the s_wait before each WMMA. Requirements: K % 4 == 0, N % 16 == 0.
// ---------------------------------------------------------------------------
#define GK 64                      // K chunk staged in LDS

__device__ __forceinline__ v8f wmma_step(const float* __restrict__ arow,
                                         const float* __restrict__ Bt,
                                         int k0, int kk, int half, int l, v8f acc) {
    const int ka = kk + 2 * half;                // this lane-half: K = ka, ka+1
    const float* ap = arow + k0 + ka;
    v2f a; a.x = ap[0]; a.y = ap[1];             // A[row][k0+ka .. +1]
    v2f b; b.x = Bt[(ka << 4) + l];              // B[k0+ka][col]
           b.y = Bt[((ka + 1) << 4) + l];        // B[k0+ka+1][col]
    return __builtin_amdgcn_wmma_f32_16x16x4_f32(
        /*neg_a=*/false, a, /*neg_b=*/false, b,
        /*c_mod=*/(short)0, acc, /*reuse_a=*/false, /*reuse_b=*/false);
}

__global__ void wmma_gemm_f32(const float* __restrict__ A, int lda, long sA,
                              const float* __restrict__ Bm, int ldb, long sB,
                              const float* __restrict__ bias,
                              float* __restrict__ C, int ldc, long sC,
                              int M, int N, int K, int act) {
    __shared__ float Bt[GK * 16];

    const int tid  = threadIdx.x;        // 0..127
    const int wav  = tid >> 5;           // 0..3
    const int lane = tid & 31;
    const int half = lane >> 4;          // 0/1
    const int l    = lane & 15;
    const int n0   = blockIdx.x * 16;
    const int m0   = blockIdx.y * 64 + wav * 16;
    const long bz  = blockIdx.z;
    A  += bz * sA;  Bm += bz * sB;  C += bz * sC;

    int rowA = m0 + l; if (rowA >= M) rowA = M - 1;  // clamp; clamped rows never stored
    const float* arow = A + (long)rowA * lda;

    v8f acc = {};
    for (int k0 = 0; k0 < K; k0 += GK) {
        const int kc = (K - k0) < GK ? (K - k0) : GK;    // multiple of 4
        // cooperative stage of B[k0..k0+kc) x [n0..n0+16) into LDS
        for (int i = tid; i < kc * 16; i += 128)
            Bt[i] = Bm[(long)(k0 + (i >> 4)) * ldb + n0 + (i & 15)];
        __syncthreads();
        // prefetch next A chunk for this lane's row
        if (k0 + GK < K)
            __builtin_prefetch(arow + k0 + GK + 2 * half, 0, 1);
        if (kc == GK) {
            // compile-time trip count: unroll so loads batch ahead of waits
#pragma unroll 4
            for (int kk = 0; kk < GK; kk += 4)
                acc = wmma_step(arow, Bt, k0, kk, half, l, acc);
        } else {
            for (int kk = 0; kk < kc; kk += 4)
                acc = wmma_step(arow, Bt, k0, kk, half, l, acc);
        }
        __syncthreads();
    }

    const int col = n0 + l;
    const float bv = bias ? bias[col] : 0.0f;
#pragma unroll
    for (int r = 0; r < 8; ++r) {
        const int row = m0 + r + 8 * half;   // C layout: VGPR r -> row r / 8+r
        if (row < M)
            C[(long)row * ldc + col] = act_apply(acc[r] + bv, act);
    }
}

// ---------------------------------------------------------------------------
// Naive GEMM for tiny / N==1 shapes (matvecs, meta projection, final proj).
// ---------------------------------------------------------------------------
__global__ void naive_gemm(const float* __restrict__ A, int lda,
                           const float* __restrict__ Bm, int ldb,
                           const float* __restrict__ bias,
                           float* __restrict__ C, int ldc,
                           int M, int N, int K, int act) {
    int t = blockIdx.x * blockDim.x + threadIdx.x;
    const int total = M * N;
    for (; t < total; t += gridDim.x * blockDim.x) {
        const int m = t / N, n = t % N;
        float acc = bias ? bias[n] : 0.0f;
        const float* a = A + (long)m * lda;
        for (int k = 0; k < K; ++k) acc += a[k] * Bm[(long)k * ldb + n];
        C[(long)m * ldc + n] = act_apply(acc, act);
    }
}

// ---------------------------------------------------------------------------
// Top-K node selection (descending, ties -> lowest index, matching lax.top_k)
// + gather of selected rows. One block per batch element.
// ---------------------------------------------------------------------------
#define TPK_T 256
__global__ void topk_gather(const float* __restrict__ scores,
                            const float* __restrict__ h,
                            float* __restrict__ hout,
                            int Nin, int Ksel, int Fd, int baseRow, int rowsTot) {
    __shared__ float s[NIN];
    __shared__ float rv[TPK_T];
    __shared__ int   ri[TPK_T];
    __shared__ int   sel;
    const int b = blockIdx.x, t = threadIdx.x;
    for (int i = t; i < Nin; i += TPK_T) s[i] = scores[(long)b * Nin + i];
    __syncthreads();
    for (int r = 0; r < Ksel; ++r) {
        float bv = -INFINITY; int bi = 0x7fffffff;
        for (int i = t; i < Nin; i += TPK_T) {
            const float v = s[i];
            if (v > bv || (v == bv && i < bi)) { bv = v; bi = i; }
        }
        rv[t] = bv; ri[t] = bi;
        __syncthreads();
        for (int st = TPK_T >> 1; st > 0; st >>= 1) {
            if (t < st) {
                const float v2 = rv[t + st]; const int i2 = ri[t + st];
                if (v2 > rv[t] || (v2 == rv[t] && i2 < ri[t])) { rv[t] = v2; ri[t] = i2; }
            }
            __syncthreads();
        }
        if (t == 0) { sel = ri[0]; s[ri[0]] = -INFINITY; }
        __syncthreads();
        const int src = sel;
        for (int f = t; f < Fd; f += TPK_T)
            hout[((long)b * rowsTot + baseRow + r) * Fd + f] =
                h[((long)b * Nin + src) * Fd + f];
        __syncthreads();
    }
}

// append meta row + cls row
__global__ void fill_tail(const float* __restrict__ meta, const float* __restrict__ cls,
                          float* __restrict__ hout, int rowsTot, int Fd) {
    const int b = blockIdx.x, t = threadIdx.x;
    hout[((long)b * rowsTot + rowsTot - 2) * Fd + t] = meta[(long)b * Fd + t];
    hout[((long)b * rowsTot + rowsTot - 1) * Fd + t] = cls[t];
}

// ---------------------------------------------------------------------------
// eij[b,i,j] = sum_d relu(ti[b,i,d] + tj[b,j,d] + tmb[b,d]) * e2w[d] + e2b
// (e1 bias is pre-folded into tmb). One block per (i,b) row.
// ---------------------------------------------------------------------------
__global__ void edge_scores(const float* __restrict__ ti, const float* __restrict__ tj,
                            const float* __restrict__ tmb, const float* __restrict__ e2w,
                            const float* __restrict__ e2b, float* __restrict__ eij, int N) {
    __shared__ float til[HID], w[HID];
    const int i = blockIdx.x, b = blockIdx.y, t = threadIdx.x;
    if (t < HID) {
        til[t] = ti[((long)b * N + i) * HID + t] + tmb[b * HID + t];
        w[t]   = e2w[t];
    }
    __syncthreads();
    const float bb = e2b[0];
    for (int j = t; j < N; j += blockDim.x) {
        const float* tjp = tj + ((long)b * N + j) * HID;
        float acc = 0.0f;
#pragma unroll
        for (int d = 0; d < HID; ++d)
            acc += fmaxf(til[d] + tjp[d], 0.0f) * w[d];
        eij[((long)b * N + i) * N + j] = acc + bb;
    }
}

// ---------------------------------------------------------------------------
// Per-row: top-k mask (kept entries keep eij value, pruned become 0 exactly as
// eij*mask in the reference), softmax over all N entries, scale by u[b,j].
// ---------------------------------------------------------------------------
#define MS_T 128
__global__ void mask_softmax(const float* __restrict__ eij, const float* __restrict__ u,
                             float* __restrict__ att, int N, int Ksel) {
    __shared__ float vals[NT + 4];
    __shared__ float work[NT + 4];
    __shared__ int   keep[NT + 4];
    __shared__ float rv[MS_T];
    __shared__ int   ri[MS_T];
    const int row = blockIdx.x;          // 0..B*N-1
    const int b   = row / N;
    const int t   = threadIdx.x;
    const float* e = eij + (long)row * N;
    for (int j = t; j < N; j += MS_T) { const float v = e[j]; vals[j] = v; work[j] = v; keep[j] = 0; }
    __syncthreads();
    for (int r = 0; r < Ksel; ++r) {
        float bv = -INFINITY; int bi = 0x7fffffff;
        for (int j = t; j < N; j += MS_T) {
            const float v = work[j];
            if (v > bv || (v == bv && j < bi)) { bv = v; bi = j; }
        }
        rv[t] = bv; ri[t] = bi; __syncthreads();
        for (int st = MS_T >> 1; st > 0; st >>= 1) {
            if (t < st) {
                const float v2 = rv[t + st]; const int i2 = ri[t + st];
                if (v2 > rv[t] || (v2 == rv[t] && i2 < ri[t])) { rv[t] = v2; ri[t] = i2; }
            }
            __syncthreads();
        }
        if (t == 0) { keep[ri[0]] = 1; work[ri[0]] = -INFINITY; }
        __syncthreads();
    }
    for (int j = t; j < N; j += MS_T) work[j] = keep[j] ? vals[j] : 0.0f;
    __syncthreads();
    float mx = -INFINITY;
    for (int j = t; j < N; j += MS_T) mx = fmaxf(mx, work[j]);
    rv[t] = mx; __syncthreads();
    for (int st = MS_T >> 1; st > 0; st >>= 1) { if (t < st) rv[t] = fmaxf(rv[t], rv[t + st]); __syncthreads(); }
    mx = rv[0]; __syncthreads();
    float sm = 0.0f;
    for (int j = t; j < N; j += MS_T) sm += expf(work[j] - mx);
    rv[t] = sm; __syncthreads();
    for (int st = MS_T >> 1; st > 0; st >>= 1) { if (t < st) rv[t] += rv[t + st]; __syncthreads(); }
    sm = rv[0]; __syncthreads();
    const float inv = 1.0f / sm;
    float* arow = att + (long)row * N;
    const float* ub = u + (long)b * N;
    for (int j = t; j < N; j += MS_T) arow[j] = expf(work[j] - mx) * inv * ub[j];
}

// residual add + LayerNorm over F=256, one block (256 threads) per row, in place
__global__ void resid_ln(const float* __restrict__ xnew, float* __restrict__ h,
                         const float* __restrict__ g, const float* __restrict__ beta) {
    __shared__ float red[F];
    const long row = blockIdx.x;
    const int  t   = threadIdx.x;
    const float x = xnew[row * F + t] + h[row * F + t];
    red[t] = x; __syncthreads();
    for (int st = F >> 1; st > 0; st >>= 1) { if (t < st) red[t] += red[t + st]; __syncthreads(); }
    const float mu = red[0] / (float)F; __syncthreads();
    const float d = x - mu;
    red[t] = d * d; __syncthreads();
    for (int st = F >> 1; st > 0; st >>= 1) { if (t < st) red[t] += red[t + st]; __syncthreads(); }
    const float var = red[0] / (float)F;
    const float inv = 1.0f / sqrtf(var + LN_EPS);
    h[row * F + t] = d * inv * g[t] + beta[t];
}

// ---------------------------------------------------------------------------
extern "C" void kernel_launch(void* const* d_in, const int* in_sizes, int n_in,
                              void* d_out, int out_size, void* d_ws, size_t ws_size,
                              hipStream_t stream) {
    (void)in_sizes; (void)n_in; (void)out_size; (void)ws_size;
    auto P = [&](int i) { return (const float*)d_in[i]; };

    // --- pytree leaf order (jax sorted-dict flatten) ---
    int idx = 0;
    const float* h1   = P(idx++);
    const float* h2   = P(idx++);
    const float* meta = P(idx++);
    const float* cls  = P(idx++);                       // params.cls
    struct HeadP { const float *W,*e1b,*e1w,*e2b,*e2w,*u1b,*u1w,*u2b,*u2w; };
    struct LayerP { const float *fcb,*fcw; HeadP hd[4]; };
    LayerP L[2];
    for (int l = 0; l < 2; ++l) {
        L[l].fcb = P(idx++); L[l].fcw = P(idx++);       // fc.b, fc.w
        for (int hh = 0; hh < 4; ++hh) {                // W, e1.{b,w}, e2.{b,w}, u1.{b,w}, u2.{b,w}
            HeadP& H = L[l].hd[hh];
            H.W   = P(idx++);
            H.e1b = P(idx++); H.e1w = P(idx++);
            H.e2b = P(idx++); H.e2w = P(idx++);
            H.u1b = P(idx++); H.u1w = P(idx++);
            H.u2b = P(idx++); H.u2w = P(idx++);
        }
    }
    const float* ln_b = P(idx++);
    const float* ln_g = P(idx++);
    const float *pfc1b[2], *pfc1w[2], *pfc2b[2], *pfc2w[2];
    for (int p = 0; p < 2; ++p) {
        pfc1b[p] = P(idx++); pfc1w[p] = P(idx++);
        pfc2b[p] = P(idx++); pfc2w[p] = P(idx++);
    }
    const float* projb = P(idx++);
    const float* projw = P(idx++);

    // --- workspace layout (floats), ~11.6 MB total ---
    float* ws      = (float*)d_ws;
    float* hbuf    = ws;                                // [B*NT, F]
    float* poolhid = hbuf    + (long)BATCH*NT*F;        // [B*NIN, F]
    float* scbuf   = poolhid + (long)BATCH*NIN*F;       // [B*NIN]
    float* Whb     = scbuf   + (long)BATCH*NIN;         // [B*NT, F]
    float* tib     = Whb     + (long)BATCH*NT*F;        // [B*NT, HID]
    float* tjb     = tib     + (long)BATCH*NT*HID;      // [B*NT, HID]
    float* tmbb    = tjb     + (long)BATCH*NT*HID;      // [B, HID]
    float* eijb    = tmbb    + (long)BATCH*HID;         // [B, NT, NT]
    float* attb    = eijb    + (long)BATCH*NT*NT;       // [B, NT, NT]
    float* uhid    = attb    + (long)BATCH*NT*NT;       // [B*NT, UH]
    float* ubuf    = uhid    + (long)BATCH*NT*UH;       // [B*NT]
    float* catb    = ubuf    + (long)BATCH*NT;          // [B*NT, 4F]
    float* tmpb    = catb    + (long)BATCH*NT*4*F;      // [B*NT, F]

    const int MT   = BATCH * NT;                 // 824 rows
    const int my64 = (MT + 63) / 64;             // 13 row panels
    dim3 blk(128);                               // 4 waves per 64x16 C panel

    // ---------------- pooling ----------------
    for (int p = 0; p < 2; ++p) {
        const float* hp = p ? h2 : h1;
        // tanh(h @ fc1.w + fc1.b)  [2048,256]
        wmma_gemm_f32<<<dim3(F/16, (BATCH*NIN)/64, 1), blk, 0, stream>>>(
            hp, F, 0, pfc1w[p], F, 0, pfc1b[p], poolhid, F, 0, BATCH*NIN, F, F, 1);
        // scores = hid @ fc2.w + fc2.b  [2048]
        naive_gemm<<<8, 256, 0, stream>>>(poolhid, F, pfc2w[p], 1, pfc2b[p],
                                          scbuf, 1, BATCH*NIN, 1, F, 0);
        // gather top-205 rows in score order
        topk_gather<<<BATCH, TPK_T, 0, stream>>>(scbuf, hp, hbuf, NIN, KPOOL, F,
                                                 p * KPOOL, NT);
    }
    fill_tail<<<BATCH, F, 0, stream>>>(meta, cls, hbuf, NT, F);

    // ---------------- GAT layers ----------------
    for (int l = 0; l < 2; ++l) {
        for (int hh = 0; hh < 4; ++hh) {
            const HeadP& H = L[l].hd[hh];
            // Wh = h @ W
            wmma_gemm_f32<<<dim3(F/16, my64, 1), blk, 0, stream>>>(
                hbuf, F, 0, H.W, F, 0, nullptr, Whb, F, 0, MT, F, F, 0);
            // ti = Wh @ e1.w[:F],  tj = Wh @ e1.w[F:2F]
            wmma_gemm_f32<<<dim3(HID/16, my64, 1), blk, 0, stream>>>(
                Whb, F, 0, H.e1w, HID, 0, nullptr, tib, HID, 0, MT, HID, F, 0);
            wmma_gemm_f32<<<dim3(HID/16, my64, 1), blk, 0, stream>>>(
                Whb, F, 0, H.e1w + (long)F*HID, HID, 0, nullptr, tjb, HID, 0, MT, HID, F, 0);
            // tmb = meta @ e1.w[2F:] + e1.b   [B,HID]
            naive_gemm<<<1, BATCH*HID, 0, stream>>>(meta, METAF, H.e1w + (long)2*F*HID, HID,
                                                    H.e1b, tmbb, HID, BATCH, HID, METAF, 0);
            // eij
            edge_scores<<<dim3(NT, BATCH), 256, 0, stream>>>(tib, tjb, tmbb, H.e2w, H.e2b,
                                                             eijb, NT);
            // u = sigmoid(relu(h@u1+b) @ u2 + b)
            wmma_gemm_f32<<<dim3(UH/16, my64, 1), blk, 0, stream>>>(
                hbuf, F, 0, H.u1w, UH, 0, H.u1b, uhid, UH, 0, MT, UH, F, 2);
            naive_gemm<<<4, 256, 0, stream>>>(uhid, UH, H.u2w, 1, H.u2b,
                                              ubuf, 1, MT, 1, UH, 4);
            // att = softmax(eij * top-k mask) * u   then  head_out = att @ Wh -> cat
            mask_softmax<<<MT, MS_T, 0, stream>>>(eijb, ubuf, attb, NT, KPR);
            wmma_gemm_f32<<<dim3(F/16, (NT + 63)/64, BATCH), blk, 0, stream>>>(
                attb, NT, (long)NT*NT, Whb, F, (long)NT*F, nullptr,
                catb + (long)hh*F, 4*F, (long)NT*4*F, NT, F, NT, 0);
        }
        // h' = elu(cat @ fc.w + fc.b); then residual + LayerNorm (in place on hbuf)
        wmma_gemm_f32<<<dim3(F/16, my64, 1), blk, 0, stream>>>(
            catb, 4*F, 0, L[l].fcw, F, 0, L[l].fcb, tmpb, F, 0, MT, F, 4*F, 3);
        resid_ln<<<MT, F, 0, stream>>>(tmpb, hbuf, ln_g, ln_b);
    }

    // ---------------- final projection: h[:, -1] @ proj.w + proj.b ----------------
    // row b of the 2-row A matrix starts at hbuf[(b*NT + NT-1)*F]  (lda = NT*F)
    naive_gemm<<<1, 256, 0, stream>>>(hbuf + (long)(NT - 1)*F, NT*F,
                                      projw, OUTF, projb, (float*)d_out, OUTF,
                                      BATCH, OUTF, F, 0);
}